// MVNProfile_25838523253357
// MI455X (gfx1250) — compile-verified
//
#include <hip/hip_runtime.h>

#define B_ROWS 16384
#define DMODEL 64
#define NPTS   1323
#define ROWS_PER_TILE 16
#define PCHUNKS 4
#define CHUNK_LEN ((NPTS + PCHUNKS - 1) / PCHUNKS)   /* 331 */

typedef float v2f __attribute__((ext_vector_type(2)));
typedef float v8f __attribute__((ext_vector_type(8)));

// Fused kernel: per 16-row tile, wave 0 computes the 16x9 param GEMM with
// V_WMMA_F32_16X16X4_F32 (f32-exact), 16 lanes derive Cholesky params into
// LDS, then all 8 waves stream the dxyz points for those 16 rows.
__global__ __launch_bounds__(256) void MVNProfile_kernel(
    const float* __restrict__ rep,   // (B, 64)
    const float* __restrict__ dxyz,  // (B, NPTS, 3)
    const float* __restrict__ Wm,    // (3, 64)
    const float* __restrict__ bm,    // (3,)
    const float* __restrict__ Ws,    // (6, 64)
    const float* __restrict__ bs,    // (6,)
    float* __restrict__ out)         // (B, NPTS)
{
  __shared__ float tile[16][17];   // raw GEMM results (16 rows x 16 cols, 9 used)
  __shared__ float prm[16][10];    // derived per-row params

  const int row0 = blockIdx.x * ROWS_PER_TILE;
  const int tid  = threadIdx.x;

  // ---- Phase 1: param GEMM on wave 0 (EXEC all-ones inside uniform branch) ----
  if (tid < 32) {
    const int lane  = tid;
    const int m     = lane & 15;          // A: row M / B: col N
    const int khalf = (lane >> 4) << 1;   // 0 or 2 within each K=4 block

    const float* arow = rep + (size_t)(row0 + m) * DMODEL + khalf;
    // Columns 9..15 of the 16-wide tile are never read, so padded lanes just
    // need *some* valid address (junk results are ignored). Clamp n into [0,9)
    // so every lane loads unconditionally -> no exec-mask branching in the loop.
    const int n = (m < 9) ? m : 0;
    const float* wrow = (n < 3) ? (Wm + n * DMODEL + khalf)
                                : (Ws + (n - 3) * DMODEL + khalf);

#if __has_builtin(__builtin_amdgcn_wmma_f32_16x16x4_f32)
    v8f c = {0.f, 0.f, 0.f, 0.f, 0.f, 0.f, 0.f, 0.f};
#pragma unroll
    for (int kk = 0; kk < 16; ++kk) {
      v2f a = *(const v2f*)(arow + kk * 4);
      v2f b = *(const v2f*)(wrow + kk * 4);
      c = __builtin_amdgcn_wmma_f32_16x16x4_f32(
              /*neg_a=*/false, a, /*neg_b=*/false, b,
              /*c_mod=*/(short)0, c, /*reuse_a=*/false, /*reuse_b=*/false);
    }
    // C layout: VGPR r, lanes 0-15 -> row r; lanes 16-31 -> row r+8; col = lane&15
    const int rbase = (lane >> 4) << 3;
#pragma unroll
    for (int r = 0; r < 8; ++r)
      tile[rbase + r][m] = c[r];
#else
    // Scalar fallback: lanes 0..15 each compute one row's 9 dot products.
    if (lane < 16) {
      const float* a = rep + (size_t)(row0 + lane) * DMODEL;
      for (int nn = 0; nn < 9; ++nn) {
        const float* w = (nn < 3) ? (Wm + nn * DMODEL) : (Ws + (nn - 3) * DMODEL);
        float acc = 0.f;
        for (int k = 0; k < DMODEL; ++k) acc = fmaf(a[k], w[k], acc);
        tile[lane][nn] = acc;
      }
    }
#endif
  }
  __syncthreads();

  // ---- Phase 2: derive per-row Cholesky params (16 lanes) ----
  if (tid < 16) {
    const int r = tid;
    float mean0 = tile[r][0] + bm[0];
    float mean1 = tile[r][1] + bm[1];
    float mean2 = tile[r][2] + bm[2];
    float s0 = 1.f / (1.f + expf(-(tile[r][3] + bs[0])));
    float s1 = 1.f / (1.f + expf(-(tile[r][4] + bs[1])));
    float s2 = 1.f / (1.f + expf(-(tile[r][5] + bs[2])));
    float s3 = 1.f / (1.f + expf(-(tile[r][6] + bs[3])));
    float s4 = 1.f / (1.f + expf(-(tile[r][7] + bs[4])));
    float s5 = 1.f / (1.f + expf(-(tile[r][8] + bs[5])));
    float L00 = log1pf(expf(s0));
    float L11 = log1pf(expf(s2));
    float L22 = log1pf(expf(s5));
    prm[r][0] = mean0;
    prm[r][1] = mean1;
    prm[r][2] = mean2;
    prm[r][3] = 1.f / L00;
    prm[r][4] = s1;              // L10
    prm[r][5] = 1.f / L11;
    prm[r][6] = s3;              // L20
    prm[r][7] = s4;              // L21
    prm[r][8] = 1.f / L22;
    // -logdet - 1.5*log(2*pi)
    prm[r][9] = -logf(L00 * L11 * L22) - 2.7568155996140182f;
  }
  __syncthreads();

  // ---- Phase 3: stream points (memory-bound: b96 loads, NT b32 stores) ----
  const int p0 = blockIdx.y * CHUNK_LEN;
  const int p1 = (p0 + CHUNK_LEN < NPTS) ? (p0 + CHUNK_LEN) : NPTS;

  for (int r = 0; r < ROWS_PER_TILE; ++r) {
    const float m0  = prm[r][0], m1  = prm[r][1], m2  = prm[r][2];
    const float i00 = prm[r][3], l10 = prm[r][4], i11 = prm[r][5];
    const float l20 = prm[r][6], l21 = prm[r][7], i22 = prm[r][8];
    const float cc  = prm[r][9];
    const float* __restrict__ q = dxyz + (size_t)(row0 + r) * (NPTS * 3);
    float* __restrict__ o       = out  + (size_t)(row0 + r) * NPTS;

    for (int pt = p0 + tid; pt < p1; pt += 256) {
      const float* qq = q + 3 * pt;       // contiguous 12B/lane -> global_load_b96
      float d0 = qq[0] - m0;
      float d1 = qq[1] - m1;
      float d2 = qq[2] - m2;
      float z0 = d0 * i00;
      float z1 = (d1 - l10 * z0) * i11;
      float z2 = (d2 - l20 * z0 - l21 * z1) * i22;
      float maha = fmaf(z0, z0, fmaf(z1, z1, z2 * z2));
      __builtin_nontemporal_store(__expf(fmaf(-0.5f, maha, cc)), o + pt);
    }
  }
}

extern "C" void kernel_launch(void* const* d_in, const int* in_sizes, int n_in,
                              void* d_out, int out_size, void* d_ws, size_t ws_size,
                              hipStream_t stream) {
  const float* rep  = (const float*)d_in[0];
  const float* dxyz = (const float*)d_in[1];
  const float* Wm   = (const float*)d_in[2];
  const float* bm   = (const float*)d_in[3];
  const float* Ws   = (const float*)d_in[4];
  const float* bs   = (const float*)d_in[5];
  // d_in[6] = num_planes (always 3) — unused.
  float* out = (float*)d_out;

  dim3 grid(B_ROWS / ROWS_PER_TILE, PCHUNKS);
  MVNProfile_kernel<<<grid, 256, 0, stream>>>(rep, dxyz, Wm, bm, bs ? Ws : Ws, bs, out);
}